// GenerativeSLDS_52432960749769
// MI455X (gfx1250) — compile-verified
//
#include <hip/hip_runtime.h>
#include <hip/hip_bf16.h>

#define NS 32
#define TT 4096
#define DD 16
#define KK 8
#define TF (NS * TT) // 131072

typedef float v2f __attribute__((ext_vector_type(2)));
typedef float v4f __attribute__((ext_vector_type(4)));
typedef float v8f __attribute__((ext_vector_type(8)));

// ---------------------------------------------------------------------------
// Prep: Sig = sqrtS @ sqrtS^T, SigInit likewise, into d_ws; zero d_out.
// ws layout: [0,2048) Sig (k,i,j) ; [2048,4096) Sig_init (k,i,j)
// ---------------------------------------------------------------------------
__global__ void slds_prep(const float* __restrict__ sqS,
                          const float* __restrict__ sqSi,
                          float* __restrict__ ws_sig,
                          float* __restrict__ out) {
  const int tid = threadIdx.x;
  if (tid < NS) out[tid] = 0.0f;
  for (int e = tid; e < KK * DD * DD; e += blockDim.x) {
    const int k = e >> 8, rem = e & 255, i = rem >> 4, j = rem & 15;
    const float* a = sqS + k * 256;
    const float* b = sqSi + k * 256;
    float s = 0.f, si = 0.f;
#pragma unroll
    for (int p = 0; p < DD; ++p) {
      s  += a[i * 16 + p] * a[j * 16 + p];
      si += b[i * 16 + p] * b[j * 16 + p];
    }
    ws_sig[e] = s;
    ws_sig[KK * DD * DD + e] = si;
  }
}

// ---------------------------------------------------------------------------
// Main: per wave, 4 tiles of 16 timesteps.
//   GEMM2 (WMMA f32 16x16x4): SigW(16x256) = Z(16x8) @ SigFlat(8x256)
//   GEMM1 (WMMA f32 16x16x4): M(16x128)    = Xprev(16x16) @ A^T(16x128)
//   then wave-cooperative 16x16 Cholesky + solve, 2 timesteps per pass.
// ---------------------------------------------------------------------------
__launch_bounds__(64)
__global__ void slds_main(const float* __restrict__ xs,
                          const float* __restrict__ zs,
                          const float* __restrict__ As,
                          const float* __restrict__ bs,
                          const float* __restrict__ mu_init,
                          const float* __restrict__ ws_sig,
                          float* __restrict__ out) {
  // Sig stored TRANSPOSED: SigT[c][k], c = i*16+j  ->  per-lane WMMA-B pairs
  // (k, k+1) are contiguous 8-byte LDS loads.
  __shared__ __align__(16) float lds_SigT[256 * KK]; //  8 KB
  __shared__ float lds_bs[128];
  __shared__ float lds_mui[128];
  __shared__ __align__(16) float lds_SigW[2][16 * 256]; // 32 KB per-wave staging
  __shared__ __align__(16) float lds_M[2][16 * 128];    // 16 KB per-wave staging

  const int tid  = threadIdx.x;
  const int lane = tid & 31;
  const int wv   = tid >> 5;
  const int hi   = lane >> 4;  // lane-half (K/M split per ISA WMMA layouts)
  const int mr   = lane & 15;  // A-row (M) / B,C,D column (N)

  for (int e = tid; e < KK * 256; e += 64) {
    const int k = e >> 8, c = e & 255;
    lds_SigT[c * KK + k] = ws_sig[e];
  }
  for (int e = tid; e < 128; e += 64) { lds_bs[e] = bs[e]; lds_mui[e] = mu_init[e]; }
  __syncthreads();

  const int B0 = blockIdx.x * 128;
  const int waveBase = B0 + wv * 64;
  const float* SigInit = ws_sig + KK * 256;

  // ---- hoist loop-invariant WMMA B-operands into registers ----------------
  v2f sigB[16][2]; // Sigma row tile ct, K-chunks {2hi,2hi+1} and {4+2hi,5+2hi}
#pragma unroll
  for (int ct = 0; ct < 16; ++ct) {
    const float* p = &lds_SigT[(ct * 16 + mr) * KK];
    sigB[ct][0] = *(const v2f*)(p + 2 * hi);
    sigB[ct][1] = *(const v2f*)(p + 4 + 2 * hi);
  }
  v2f aB[KK][4]; // As[k] row d=mr, e-chunks of 2
#pragma unroll
  for (int k = 0; k < KK; ++k) {
    const float* Ab = As + k * 256 + mr * 16;
#pragma unroll
    for (int c = 0; c < 4; ++c) aB[k][c] = *(const v2f*)(Ab + 4 * c + 2 * hi);
  }

  float lpAcc = 0.f;

  for (int it = 0; it < 4; ++it) {
    const int f0 = waveBase + it * 16;

    // ---------------- GEMM2: weighted Sigma, K-dim = 8 (2 chunks) ----------
    {
      const float* zr = zs + (size_t)(f0 + mr) * KK;
      v2f za0 = *(const v2f*)(zr + 2 * hi);
      v2f za1 = *(const v2f*)(zr + 4 + 2 * hi);
#pragma unroll
      for (int ct = 0; ct < 16; ++ct) { // Sigma row i = ct
        v8f acc = {0.f, 0.f, 0.f, 0.f, 0.f, 0.f, 0.f, 0.f};
        acc = __builtin_amdgcn_wmma_f32_16x16x4_f32(false, za0, false, sigB[ct][0],
                                                    (short)0, acc, false, false);
        acc = __builtin_amdgcn_wmma_f32_16x16x4_f32(false, za1, false, sigB[ct][1],
                                                    (short)0, acc, false, false);
#pragma unroll
        for (int r = 0; r < 8; ++r)
          lds_SigW[wv][(r + 8 * hi) * 256 + ct * 16 + mr] = acc[r];
      }
    }

    // ---------------- GEMM1: AR means, K-dim = 16 (4 chunks) ---------------
    {
      int fp = f0 + mr - 1;
      if (fp < 0) fp = 0; // f==0 mu overridden by mu_init later
      const float* xr = xs + (size_t)fp * DD;
      v2f xa0 = *(const v2f*)(xr + 0 + 2 * hi);
      v2f xa1 = *(const v2f*)(xr + 4 + 2 * hi);
      v2f xa2 = *(const v2f*)(xr + 8 + 2 * hi);
      v2f xa3 = *(const v2f*)(xr + 12 + 2 * hi);
#pragma unroll
      for (int k = 0; k < KK; ++k) {
        v8f acc = {0.f, 0.f, 0.f, 0.f, 0.f, 0.f, 0.f, 0.f};
        acc = __builtin_amdgcn_wmma_f32_16x16x4_f32(false, xa0, false, aB[k][0],
                                                    (short)0, acc, false, false);
        acc = __builtin_amdgcn_wmma_f32_16x16x4_f32(false, xa1, false, aB[k][1],
                                                    (short)0, acc, false, false);
        acc = __builtin_amdgcn_wmma_f32_16x16x4_f32(false, xa2, false, aB[k][2],
                                                    (short)0, acc, false, false);
        acc = __builtin_amdgcn_wmma_f32_16x16x4_f32(false, xa3, false, aB[k][3],
                                                    (short)0, acc, false, false);
#pragma unroll
        for (int r = 0; r < 8; ++r)
          lds_M[wv][(r + 8 * hi) * 128 + k * 16 + mr] = acc[r];
      }
    }

    // --------- boundary fix-up: f % 4096 == 0 uses Sig_init ----------------
    if (wv == 0 && it == 0 && (B0 & (TT - 1)) == 0) {
      const float* z0 = zs + (size_t)B0 * KK;
      float zk[KK];
#pragma unroll
      for (int k = 0; k < KK; ++k) zk[k] = z0[k];
      for (int c = lane; c < 256; c += 32) {
        float s = 0.f;
#pragma unroll
        for (int k = 0; k < KK; ++k) s += zk[k] * SigInit[k * 256 + c];
        lds_SigW[0][c] = s; // row t = 0
      }
    }

    __syncthreads();

    // ---------------- Cholesky + solve: 8 passes, 2 timesteps each ---------
    for (int p = 0; p < 8; ++p) {
      const int t = 2 * p + hi;
      const int f = f0 + t;
      const v4f* sw4 = (const v4f*)&lds_SigW[wv][t * 256 + mr * 16];
      v4f q0 = sw4[0], q1 = sw4[1], q2 = sw4[2], q3 = sw4[3]; // ds_load_b128 x4
      float L[16] = {q0.x, q0.y, q0.z, q0.w, q1.x, q1.y, q1.z, q1.w,
                     q2.x, q2.y, q2.z, q2.w, q3.x, q3.y, q3.z, q3.w};

      // residual r = x[f] - mu_w[f]
      const float* zf = zs + (size_t)f * KK;
      float zk[KK];
#pragma unroll
      for (int k = 0; k < KK; ++k) zk[k] = zf[k];
      float mu = 0.f;
      if (f == 0) {
#pragma unroll
        for (int k = 0; k < KK; ++k) mu += zk[k] * lds_mui[k * 16 + mr];
      } else {
#pragma unroll
        for (int k = 0; k < KK; ++k)
          mu += zk[k] * (lds_bs[k * 16 + mr] + lds_M[wv][t * 128 + k * 16 + mr]);
      }
      float r = xs[(size_t)f * DD + mr] - mu;

      const int base = hi << 4;
      float myInvDiag = 1.f, ldsum = 0.f;
#pragma unroll
      for (int j = 0; j < 16; ++j) {
        float s = L[j];
#pragma unroll
        for (int q = 0; q < j; ++q) {
          float Ljq = __shfl(L[q], base + j, 32);
          s -= L[q] * Ljq;
        }
        float dj  = __shfl(s, base + j, 32);
        float Ljj = sqrtf(dj);
        float inv = 1.0f / Ljj;
        ldsum += __logf(Ljj);
        L[j] = (mr == j) ? Ljj : s * inv;
        if (mr == j) myInvDiag = inv;
      }
      // forward substitution L y = r; mahal = |y|^2
      float acc2 = 0.f, msum = 0.f;
#pragma unroll
      for (int j = 0; j < 16; ++j) {
        float v  = (r - acc2) * myInvDiag;
        float yj = __shfl(v, base + j, 32);
        msum += yj * yj;
        acc2 += L[j] * yj;
      }
      const float lp = -0.5f * (29.4060330626f + 2.f * ldsum + msum); // D*log(2pi)+...
      if (mr == 0) lpAcc += lp;
    }
    __syncthreads();
  }

  // combine half-wave accumulators and add into out[n]
  float other = __shfl(lpAcc, lane ^ 16, 32);
  if (lane == 0) {
    const int n = waveBase >> 12; // / 4096
    atomicAdd(&out[n], lpAcc + other);
  }
}

// ---------------------------------------------------------------------------
extern "C" void kernel_launch(void* const* d_in, const int* in_sizes, int n_in,
                              void* d_out, int out_size, void* d_ws, size_t ws_size,
                              hipStream_t stream) {
  (void)in_sizes; (void)n_in; (void)out_size; (void)ws_size;
  const float* xs   = (const float*)d_in[0];
  const float* zs   = (const float*)d_in[1];
  const float* As   = (const float*)d_in[2];
  const float* bs   = (const float*)d_in[3];
  const float* mui  = (const float*)d_in[4];
  const float* sqSi = (const float*)d_in[5];
  const float* sqS  = (const float*)d_in[6];
  float* out = (float*)d_out;
  float* ws  = (float*)d_ws;

  slds_prep<<<1, 256, 0, stream>>>(sqS, sqSi, ws, out);
  slds_main<<<TF / 128, 64, 0, stream>>>(xs, zs, As, bs, mui, ws, out);
}